// Linear_act_sp_84859963835100
// MI455X (gfx1250) — compile-verified
//
#include <hip/hip_runtime.h>

typedef __attribute__((ext_vector_type(4)))  float    v4f;
typedef __attribute__((ext_vector_type(8)))  float    v8f;
typedef __attribute__((ext_vector_type(8)))  _Float16 v8h;
typedef __attribute__((ext_vector_type(16))) _Float16 v16h;
typedef __attribute__((ext_vector_type(32))) _Float16 v32h;

#define M_TOT 16384
#define K_TOT 4096
#define N_TOT 1024
#define KC    (K_TOT / 2)    // compressed f16 per row
#define IDXW  (K_TOT / 32)   // u32 index words per row (8 groups of 4 per word)

// ---------------------------------------------------------------------------
// Pass 1: 2:4 magnitude prune + compress. One thread = 32 expanded values
// (8 groups of 4) = 16 compressed f16 + 1 index word.
// Index word bit layout matches V_SWMMAC 16-bit index VGPR: group g -> bits
// [4g+1:4g]=idx0, [4g+3:4g+2]=idx1, with idx0 < idx1.
// ---------------------------------------------------------------------------
__device__ __forceinline__ float pick4(int i, float x0, float x1, float x2, float x3) {
  float r = x0;
  r = (i == 1) ? x1 : r;
  r = (i == 2) ? x2 : r;
  r = (i == 3) ? x3 : r;
  return r;
}

__global__ void __launch_bounds__(256)
prune_compress_kernel(const float* __restrict__ x,
                      _Float16* __restrict__ ac,
                      unsigned* __restrict__ aidx) {
  const int tid = blockIdx.x * 256 + threadIdx.x;
  const int row = tid >> 7;      // / IDXW
  const int w   = tid & (IDXW - 1);
  const float* xp = x + (size_t)row * K_TOT + w * 32;

  union { _Float16 e[16]; v8h h[2]; } o;
  unsigned idxword = 0;

#pragma unroll
  for (int g = 0; g < 8; ++g) {
    v4f v = *(const v4f*)(xp + 4 * g);
    float x0 = v[0], x1 = v[1], x2 = v[2], x3 = v[3];
    float a0 = __builtin_fabsf(x0), a1 = __builtin_fabsf(x1);
    float a2 = __builtin_fabsf(x2), a3 = __builtin_fabsf(x3);
    // first argmax (strict > keeps lowest index on ties, matching lax.top_k)
    int m1 = 0; float b1 = a0;
    if (a1 > b1) { b1 = a1; m1 = 1; }
    if (a2 > b1) { b1 = a2; m1 = 2; }
    if (a3 > b1) { b1 = a3; m1 = 3; }
    // second argmax over the remaining three
    int m2 = -1; float b2 = -1.0f;
    if (m1 != 0 && a0 > b2) { b2 = a0; m2 = 0; }
    if (m1 != 1 && a1 > b2) { b2 = a1; m2 = 1; }
    if (m1 != 2 && a2 > b2) { b2 = a2; m2 = 2; }
    if (m1 != 3 && a3 > b2) { b2 = a3; m2 = 3; }
    int i0 = m1 < m2 ? m1 : m2;
    int i1 = m1 < m2 ? m2 : m1;
    o.e[2 * g]     = (_Float16)pick4(i0, x0, x1, x2, x3);
    o.e[2 * g + 1] = (_Float16)pick4(i1, x0, x1, x2, x3);
    idxword |= (unsigned)(i0 | (i1 << 2)) << (4 * g);
  }

  _Float16* op = ac + (size_t)row * KC + w * 16;
  *(v8h*)(op)     = o.h[0];
  *(v8h*)(op + 8) = o.h[1];
  aidx[(size_t)row * IDXW + w] = idxword;
}

// ---------------------------------------------------------------------------
// Pass 2: weight fp32 -> f16 (row-major [N, K]); result is 8.4 MB, L2-resident.
// ---------------------------------------------------------------------------
__global__ void __launch_bounds__(256)
convert_w_kernel(const float* __restrict__ w, _Float16* __restrict__ wh) {
  const int tid = blockIdx.x * 256 + threadIdx.x;  // one per 8 elements
  const float* p = w + (size_t)tid * 8;
  v4f p0 = *(const v4f*)p;
  v4f p1 = *(const v4f*)(p + 4);
  v8h o;
#pragma unroll
  for (int i = 0; i < 4; ++i) { o[i] = (_Float16)p0[i]; o[i + 4] = (_Float16)p1[i]; }
  *(v8h*)(wh + (size_t)tid * 8) = o;
}

// ---------------------------------------------------------------------------
// Pass 3: sparse GEMM with V_SWMMAC_F32_16X16X64_F16.
// Block = 256 threads = 8 waves, arranged 4 (M) x 2 (N). Wave tile: 16M x 64N.
// Block tile: 64M x 128N. Grid: (M/64) x (N/128).
// Operand VGPR layouts per CDNA5 ISA 7.12.2 / 7.12.4:
//  A (16x32 compressed f16): lane<16 -> row=lane, kc {0..7,16..23}+t*32
//                            lane>=16 -> row=lane-16, kc {8..15,24..31}+t*32
//  index: lane<16 -> word covering expanded K 0..31; lane>=16 -> K 32..63
//  B (64x16 f16): col = lane&15; lane<16 -> K {0..15, 32..47}; lane>=16 -> +16
// ---------------------------------------------------------------------------
__global__ void __launch_bounds__(256)
swmmac_gemm_kernel(const _Float16* __restrict__ ac,
                   const unsigned* __restrict__ aidx,
                   const _Float16* __restrict__ wh,
                   float* __restrict__ out) {
  const int lane   = threadIdx.x & 31;
  const int wave   = threadIdx.x >> 5;
  const int wave_m = wave >> 1;
  const int wave_n = wave & 1;
  const int m0 = blockIdx.x * 64 + wave_m * 16;
  const int n0 = blockIdx.y * 128 + wave_n * 64;
  const int half = lane >> 4;
  const int lr   = lane & 15;

  const int arow = m0 + lr;
  const _Float16* abase = ac + (size_t)arow * KC + half * 8;
  const unsigned* ibase = aidx + (size_t)arow * IDXW + half;

  v8f acc[4] = {};

  for (int t = 0; t < K_TOT / 64; ++t) {
    union { v16h v; v8h h[2]; } A;
    const _Float16* ap = abase + t * 32;
    A.h[0] = *(const v8h*)(ap);
    A.h[1] = *(const v8h*)(ap + 16);
    const int iw = (int)ibase[t * 2];

#pragma unroll
    for (int s = 0; s < 4; ++s) {
      union { v32h v; v8h h[4]; } B;
      const _Float16* bp = wh + (size_t)(n0 + s * 16 + lr) * K_TOT + t * 64 + half * 16;
      B.h[0] = *(const v8h*)(bp);
      B.h[1] = *(const v8h*)(bp + 8);
      B.h[2] = *(const v8h*)(bp + 32);
      B.h[3] = *(const v8h*)(bp + 40);
      acc[s] = __builtin_amdgcn_swmmac_f32_16x16x64_f16(
          /*neg_a=*/false, A.v, /*neg_b=*/false, B.v,
          acc[s], iw, /*reuse_a=*/false, /*reuse_b=*/false);
    }
  }

  // Epilogue: 16x16 f32 C/D layout -> VGPR r holds M=r+8*half, N=lr.
#pragma unroll
  for (int s = 0; s < 4; ++s) {
    float* op = out + (size_t)(m0 + half * 8) * N_TOT + n0 + s * 16 + lr;
#pragma unroll
    for (int r = 0; r < 8; ++r)
      op[(size_t)r * N_TOT] = acc[s][r];
  }
}

extern "C" void kernel_launch(void* const* d_in, const int* in_sizes, int n_in,
                              void* d_out, int out_size, void* d_ws, size_t ws_size,
                              hipStream_t stream) {
  const float* x = (const float*)d_in[0];    // [4, 4096, 4096]
  const float* w = (const float*)d_in[1];    // [1024, 4096]
  float* out = (float*)d_out;                // [16384, 1024]

  char* ws = (char*)d_ws;
  _Float16* ac   = (_Float16*)ws;                                       // 64 MB
  unsigned* aidx = (unsigned*)(ws + (size_t)M_TOT * KC * 2);            //  8 MB
  _Float16* wh   = (_Float16*)(ws + (size_t)M_TOT * KC * 2
                                  + (size_t)M_TOT * IDXW * 4);          //  8 MB

  prune_compress_kernel<<<(M_TOT * IDXW) / 256, 256, 0, stream>>>(x, ac, aidx);
  convert_w_kernel<<<((size_t)N_TOT * K_TOT / 8) / 256, 256, 0, stream>>>(w, wh);

  dim3 grid(M_TOT / 64, N_TOT / 128);
  swmmac_gemm_kernel<<<grid, 256, 0, stream>>>(ac, aidx, wh, out);
}